// GTLayer_76441827934665
// MI455X (gfx1250) — compile-verified
//
#include <hip/hip_runtime.h>

#define DIMH 128
#define HEADS 8

typedef __attribute__((ext_vector_type(2))) float        v2f;
typedef __attribute__((ext_vector_type(8))) float        v8f;
typedef __attribute__((ext_vector_type(4))) unsigned int v4u;
typedef __attribute__((ext_vector_type(8))) int          v8i;
typedef __attribute__((ext_vector_type(4))) int          v4i;

struct MatOut {
  const float* W;   // K x NOUT, row-major
  const float* B;   // NOUT bias
  float*       O;   // nrows x NOUT output
  const float* R;   // optional residual nrows x NOUT (nullptr if none)
};

// ---------------------------------------------------------------------------
// Tensor Data Mover: async 2D tile load (rows_tile x width_elems fp32, row
// stride = stride_elems) from global into LDS.  tensor_dim1 = rows_avail so
// tail rows past the end of the tensor are zero-filled by the TDM's OOB rule.
// Issued by one wave; completion via TENSORcnt.
// ---------------------------------------------------------------------------
__device__ inline void tdm_load_2d(void* lds_dst, const void* gsrc,
                                   int width_elems, int rows_tile,
                                   int rows_avail, int stride_elems)
{
  unsigned long long ga = (unsigned long long)(size_t)gsrc;
  v4u g0;
  g0[0] = 1u;                                   // count=1, user descriptor
  g0[1] = (unsigned)(size_t)lds_dst;            // lds_addr (bytes)
  g0[2] = (unsigned)(ga & 0xFFFFFFFFull);       // global_addr[31:0]
  g0[3] = (unsigned)((ga >> 32) & 0x1FFFFFFull) // global_addr[56:32]
          | (2u << 30);                         // type=2 ("image")

  v8i g1;
  g1[0] = (2 << 16);                                        // data_size=4B
  g1[1] = (int)((width_elems & 0xFFFF) << 16);              // tensor_dim0 lo16
  g1[2] = (int)(((unsigned)width_elems >> 16) & 0xFFFF)     // tensor_dim0 hi16
          | (int)((rows_avail & 0xFFFF) << 16);             // tensor_dim1 lo16
  g1[3] = (int)(((unsigned)rows_avail >> 16) & 0xFFFF)      // tensor_dim1 hi16
          | (int)((width_elems & 0xFFFF) << 16);            // tile_dim0
  g1[4] = (rows_tile & 0xFFFF);                             // tile_dim1 (tile_dim2=0)
  g1[5] = stride_elems;                                     // tensor_dim0_stride[31:0]
  g1[6] = 0;                                                // stride0 hi / stride1 lo
  g1[7] = 0;                                                // stride1 hi

  v4i zz4 = {0, 0, 0, 0};
  v8i zz8 = {0, 0, 0, 0, 0, 0, 0, 0};
  __builtin_amdgcn_tensor_load_to_lds(g0, g1, zz4, zz4, zz8, 0);
}

// ---------------------------------------------------------------------------
// Fused [TDM stage (+LN / +softmax-normalize)] -> WMMA GEMM -> epilogue.
// One 16-row tile per wave; A tile in LDS; B fragments from global (L2-hot).
// AMODE: 0 = raw, 1 = LayerNorm, 2 = divide by per-head softmax sum.
// ---------------------------------------------------------------------------
template<int ROWS, int K, int NOUT, int AMODE, bool RELU>
__global__ __launch_bounds__((ROWS / 16) * 32)
void gemm_fused(const float* __restrict__ Araw,
                const float* __restrict__ lng, const float* __restrict__ lnb,
                const float* __restrict__ ssum,
                MatOut m0, MatOut m1, MatOut m2, int nmats,
                int nrows)
{
  __shared__ float sA[ROWS * K];
  __shared__ float sg[K];
  __shared__ float sb[K];
  const int tid  = threadIdx.x;
  const int nthr = (ROWS / 16) * 32;
  const int r0   = blockIdx.x * ROWS;

  // Async DMA of the activation tile into LDS (zero wave instructions per
  // element; tail rows zero-filled via tensor_dim1 OOB rule).
  if (tid < 32) {
    tdm_load_2d(sA, Araw + (size_t)r0 * K, K, ROWS, nrows - r0, K);
  }

  // Speculative prefetch of the next block's tile (global_prefetch_b8).
  __builtin_prefetch(Araw + (size_t)(r0 + ROWS) * K, 0, 1);

  if (AMODE == 1) {
    for (int i = tid; i < K; i += nthr) { sg[i] = lng[i]; sb[i] = lnb[i]; }
  }

  if (tid < 32) __builtin_amdgcn_s_wait_tensorcnt(0);
  __syncthreads();

  if (AMODE == 1) {  // in-place LayerNorm, one row per thread
    if (tid < ROWS && (r0 + tid) < nrows) {
      float* row = &sA[tid * K];
      float mu = 0.f;
      #pragma unroll 4
      for (int c = 0; c < K; ++c) mu += row[c];
      mu *= (1.0f / K);
      float var = 0.f;
      #pragma unroll 4
      for (int c = 0; c < K; ++c) { float d = row[c] - mu; var += d * d; }
      var *= (1.0f / K);
      float rs = rsqrtf(var + 1e-5f);
      for (int c = 0; c < K; ++c) row[c] = (row[c] - mu) * rs * sg[c] + sb[c];
    }
    __syncthreads();
  }

  if (AMODE == 2) {  // scale by 1/softmax-sum (per dest node, per head)
    for (int idx = tid * 4; idx < ROWS * K; idx += nthr * 4) {
      int row  = idx / K;
      int col  = idx - row * K;
      int grow = r0 + row;
      if (grow < nrows) {
        float s   = ssum[(size_t)grow * HEADS + (col >> 4)];
        float inv = (s != 0.f) ? 1.0f / s : 0.f;
        float4 v4 = *(float4*)(&sA[idx]);
        v4.x *= inv; v4.y *= inv; v4.z *= inv; v4.w *= inv;
        *(float4*)(&sA[idx]) = v4;
      }
    }
    __syncthreads();
  }

  // WMMA: wave handles rows [wave*16, wave*16+16) of this block's tile.
  const int lane  = tid & 31;
  const int wave  = tid >> 5;
  const float* Aw = &sA[wave * 16 * K];
  const int mrow  = lane & 15;            // A row / C column (per ISA layout)
  const int kh    = (lane >> 4) << 1;     // K sub-offset: lanes>=16 hold K+2,K+3
  const int rbase = r0 + wave * 16 + ((lane >> 4) << 3);  // C: M = vgpr + 8*(lane/16)

  for (int mi = 0; mi < nmats; ++mi) {
    MatOut mo = (mi == 0) ? m0 : (mi == 1 ? m1 : m2);
    const float* __restrict__ W = mo.W;
    for (int ct = 0; ct < NOUT / 16; ++ct) {
      int col = (ct << 4) + mrow;
      v8f acc = {0.f, 0.f, 0.f, 0.f, 0.f, 0.f, 0.f, 0.f};
      #pragma unroll
      for (int kk = 0; kk < K; kk += 4) {
        v2f a, b;
        a.x = Aw[mrow * K + kk + kh];
        a.y = Aw[mrow * K + kk + kh + 1];
        b.x = W[(size_t)(kk + kh) * NOUT + col];
        b.y = W[(size_t)(kk + kh + 1) * NOUT + col];
        acc = __builtin_amdgcn_wmma_f32_16x16x4_f32(
            /*neg_a=*/false, a, /*neg_b=*/false, b,
            /*c_mod=*/(short)0, acc, /*reuse_a=*/false, /*reuse_b=*/false);
      }
      float bias = mo.B[col];
      #pragma unroll
      for (int r = 0; r < 8; ++r) {
        int row = rbase + r;
        if (row < nrows) {
          float y = acc[r] + bias;
          if (RELU) y = fmaxf(y, 0.f);
          if (mo.R) y += mo.R[(size_t)row * NOUT + col];
          mo.O[(size_t)row * NOUT + col] = y;
        }
      }
    }
  }
}

// ---------------------------------------------------------------------------
// Edge scatter pass: one wave per edge. Computes s_elt, per-head scores via
// quad shuffles, exp (max-pass elided: scores clipped to +-5), sigmoid gate,
// and atomically accumulates exp*v*gate and the per-head normalizer.
// ---------------------------------------------------------------------------
__global__ __launch_bounds__(256)
void edge_attn(const float* __restrict__ q, const float* __restrict__ k,
               const float* __restrict__ v, const float* __restrict__ eaP,
               const float* __restrict__ egP,
               const int* __restrict__ srcI, const int* __restrict__ dstI,
               float* __restrict__ s_elt, float* __restrict__ accum,
               float* __restrict__ ssum, int E)
{
  const int wave = threadIdx.x >> 5;
  const int lane = threadIdx.x & 31;
  const long e = (long)blockIdx.x * 8 + wave;
  if (e >= E) return;
  const int sn = srcI[e], dn = dstI[e];
  const size_t off = (size_t)e * DIMH + lane * 4;

  float4 q4 = *(const float4*)(q + (size_t)dn * DIMH + lane * 4);
  float4 k4 = *(const float4*)(k + (size_t)sn * DIMH + lane * 4);
  float4 v4 = *(const float4*)(v + (size_t)sn * DIMH + lane * 4);
  float4 a4 = *(const float4*)(eaP + off);
  float4 g4 = *(const float4*)(egP + off);

  float4 s4;
  s4.x = q4.x * k4.x + a4.x;
  s4.y = q4.y * k4.y + a4.y;
  s4.z = q4.z * k4.z + a4.z;
  s4.w = q4.w * k4.w + a4.w;
  *(float4*)(s_elt + off) = s4;

  // per-head score: sum 16 D-values spread over 4 consecutive lanes
  float p = s4.x + s4.y + s4.z + s4.w;
  p += __shfl_xor(p, 1, 32);
  p += __shfl_xor(p, 2, 32);
  float score = fminf(5.0f, fmaxf(-5.0f, p * 0.25f));  // /sqrt(16), clip
  float ex = __expf(score);                            // safe: |score|<=5

  float4 gate;
  gate.x = 1.0f / (1.0f + __expf(-g4.x));
  gate.y = 1.0f / (1.0f + __expf(-g4.y));
  gate.z = 1.0f / (1.0f + __expf(-g4.z));
  gate.w = 1.0f / (1.0f + __expf(-g4.w));

  float* ab = accum + (size_t)dn * DIMH + lane * 4;
  atomicAdd(ab + 0, ex * v4.x * gate.x);
  atomicAdd(ab + 1, ex * v4.y * gate.y);
  atomicAdd(ab + 2, ex * v4.z * gate.z);
  atomicAdd(ab + 3, ex * v4.w * gate.w);
  if ((lane & 3) == 0)
    atomicAdd(ssum + (size_t)dn * HEADS + (lane >> 2), ex);
}

// ---------------------------------------------------------------------------
extern "C" void kernel_launch(void* const* d_in, const int* in_sizes, int n_in,
                              void* d_out, int out_size, void* d_ws, size_t ws_size,
                              hipStream_t stream)
{
  const float* x    = (const float*)d_in[0];
  const float* ef   = (const float*)d_in[1];
  const float* Wq   = (const float*)d_in[2];
  const float* bq   = (const float*)d_in[3];
  const float* Wk   = (const float*)d_in[4];
  const float* bk   = (const float*)d_in[5];
  const float* Wv   = (const float*)d_in[6];
  const float* bv   = (const float*)d_in[7];
  const float* We   = (const float*)d_in[8];
  const float* be   = (const float*)d_in[9];
  const float* Wg   = (const float*)d_in[10];
  const float* bg   = (const float*)d_in[11];
  const float* Wo   = (const float*)d_in[12];
  const float* bo   = (const float*)d_in[13];
  const float* Woe  = (const float*)d_in[14];
  const float* boe  = (const float*)d_in[15];
  const float* ln1g = (const float*)d_in[16];
  const float* ln1b = (const float*)d_in[17];
  const float* ln1eg= (const float*)d_in[18];
  const float* ln1eb= (const float*)d_in[19];
  const float* ln2g = (const float*)d_in[20];
  const float* ln2b = (const float*)d_in[21];
  const float* W1   = (const float*)d_in[22];
  const float* b1   = (const float*)d_in[23];
  const float* W2   = (const float*)d_in[24];
  const float* b2   = (const float*)d_in[25];
  const int*   ei   = (const int*)d_in[26];

  const int N = in_sizes[0] / DIMH;
  const int E = in_sizes[1] / DIMH;
  const int* srcI = ei;
  const int* dstI = ei + E;

  // workspace layout (floats)
  float* ws    = (float*)d_ws;
  float* qB    = ws;
  float* kB    = qB   + (size_t)N * DIMH;
  float* vB    = kB   + (size_t)N * DIMH;
  float* eaB   = vB   + (size_t)N * DIMH;
  float* egB   = eaB  + (size_t)E * DIMH;
  float* seltB = egB  + (size_t)E * DIMH;
  float* accum = seltB+ (size_t)E * DIMH;
  float* ssumB = accum+ (size_t)N * DIMH;
  float* xmid  = ssumB+ (size_t)N * HEADS;
  float* uB    = xmid + (size_t)N * DIMH;

  float* outX = (float*)d_out;
  float* outE = outX + (size_t)N * DIMH;

  MatOut z{nullptr, nullptr, nullptr, nullptr};

  // 1) LN(x) -> q, k, v
  {
    MatOut mq{Wq, bq, qB, nullptr}, mk{Wk, bk, kB, nullptr}, mv{Wv, bv, vB, nullptr};
    gemm_fused<64, 128, 128, 1, false><<<(N + 63) / 64, 128, 0, stream>>>(
        x, ln1g, ln1b, nullptr, mq, mk, mv, 3, N);
  }
  // 2) LN(e) -> ea, eg
  {
    MatOut me{We, be, eaB, nullptr}, mg{Wg, bg, egB, nullptr};
    gemm_fused<64, 128, 128, 1, false><<<(E + 63) / 64, 128, 0, stream>>>(
        ef, ln1eg, ln1eb, nullptr, me, mg, z, 2, E);
  }
  // 3) zero accumulators, then edge scatter pass
  (void)hipMemsetAsync(accum, 0, sizeof(float) * ((size_t)N * DIMH + (size_t)N * HEADS), stream);
  edge_attn<<<(E + 7) / 8, 256, 0, stream>>>(qB, kB, vB, eaB, egB, srcI, dstI,
                                             seltB, accum, ssumB, E);
  // 4) (accum/ssum) @ Wo + x  -> x_mid
  {
    MatOut mo{Wo, bo, xmid, x};
    gemm_fused<64, 128, 128, 2, false><<<(N + 63) / 64, 128, 0, stream>>>(
        accum, nullptr, nullptr, ssumB, mo, z, z, 1, N);
  }
  // 5) relu(LN(x_mid) @ W1 + b1) -> u
  {
    MatOut mf{W1, b1, uB, nullptr};
    gemm_fused<64, 128, 256, 1, true><<<(N + 63) / 64, 128, 0, stream>>>(
        xmid, ln2g, ln2b, nullptr, mf, z, z, 1, N);
  }
  // 6) u @ W2 + b2 + x_mid -> x_out
  {
    MatOut mf{W2, b2, outX, xmid};
    gemm_fused<32, 256, 128, 0, false><<<(N + 31) / 32, 64, 0, stream>>>(
        uB, nullptr, nullptr, nullptr, mf, z, z, 1, N);
  }
  // 7) s_elt @ Woe + boe + e_in -> e_out
  {
    MatOut mf{Woe, boe, outE, ef};
    gemm_fused<64, 128, 128, 0, false><<<(E + 63) / 64, 128, 0, stream>>>(
        seltB, nullptr, nullptr, nullptr, mf, z, z, 1, E);
  }
}